// Detector_36137854829223
// MI455X (gfx1250) — compile-verified
//
#include <hip/hip_runtime.h>
#include <hip/hip_bf16.h>
#include <stdint.h>

// ---------------- problem constants ----------------
#define BATCH 8
#define HH 640
#define WW 640
#define HWN (HH * WW)          // 409600
#define TOPK 1024
#define NBIN 4096
#define CAP 4096               // candidate cap
#define TILE_F 4096            // floats staged per LDS tile (16KB)
#define SCORE_TH 0.8f
#define BOX_TH 0.1f
#define NMS_TH 0.2f

typedef __attribute__((ext_vector_type(2))) float v2f;
typedef __attribute__((ext_vector_type(8))) float v8f;
typedef __attribute__((ext_vector_type(4))) int v4i;

// ---------------- CDNA5 feature detection ----------------
#if defined(__has_builtin)
#if __has_builtin(__builtin_amdgcn_global_load_async_to_lds_b128)
#define HAVE_ASYNC_LDS 1
#endif
#if __has_builtin(__builtin_amdgcn_wmma_f32_16x16x4_f32)
#define HAVE_WMMA_F32 1
#endif
#endif

// Stage 16 bytes/lane from global into LDS. Async path uses the gfx1250
// async-to-LDS DMA tracked by ASYNCcnt; fallback is a plain vector copy.
__device__ __forceinline__ void stage16(const float* __restrict__ g, float* l) {
#if defined(HAVE_ASYNC_LDS)
  __builtin_amdgcn_global_load_async_to_lds_b128(
      (__attribute__((address_space(1))) v4i*)g,
      (__attribute__((address_space(3))) v4i*)l, 0, 0);
#else
  float4 v = *(const float4*)g;
  *(float4*)l = v;
#endif
}

__device__ __forceinline__ void stage_wait() {
#if defined(HAVE_ASYNC_LDS)
  asm volatile("s_wait_asynccnt 0" ::: "memory");
#endif
}

// =====================================================================
// Kernel 1: exact top-K per batch.
//  Phase A: histogram (4096 monotone bins over (0.8,1.0]) from LDS-staged tiles
//  Phase B: suffix-count cutoff bin c* (smallest superset >= K)
//  Phase C: compact all candidates with bin >= c* into LDS keys
//  Phase D: bitonic sort keys descending (score desc, idx asc via ~idx)
//  Phase E: emit top-1024 (idx, score)
// =====================================================================
__global__ __launch_bounds__(1024) void select_topk_kernel(
    const float* __restrict__ score, unsigned int* __restrict__ topIdx,
    float* __restrict__ topScore) {
  const int b = blockIdx.x;
  const int tid = threadIdx.x;
  __shared__ float tile[TILE_F];                      // 16 KB
  __shared__ unsigned long long keys[CAP];            // 32 KB (aliases hist)
  __shared__ int s_cut, s_cnt;
  unsigned int* hist = (unsigned int*)keys;
  const float* sb = score + (size_t)b * HWN;

  for (int i = tid; i < NBIN; i += 1024) hist[i] = 0u;
  __syncthreads();

  // Phase A: histogram
  for (int t0 = 0; t0 < HWN; t0 += TILE_F) {
    stage16(sb + t0 + tid * 4, &tile[tid * 4]);
    stage_wait();
    __syncthreads();
#pragma unroll
    for (int j = 0; j < 4; ++j) {
      float s = tile[tid * 4 + j];
      if (s > SCORE_TH) {
        int bin = (int)((s - SCORE_TH) * 20480.0f);   // 0.2 range -> 4096 bins
        bin = bin > (NBIN - 1) ? (NBIN - 1) : bin;
        atomicAdd(&hist[bin], 1u);
      }
    }
    __syncthreads();
  }

  // Phase B: cutoff bin (serial over 4096 bins: negligible)
  if (tid == 0) {
    unsigned int acc = 0;
    int cut = 0;
    for (int i = NBIN - 1; i >= 0; --i) {
      acc += hist[i];
      if (acc >= TOPK) { cut = i; break; }
    }
    s_cut = cut;
    s_cnt = 0;
  }
  __syncthreads();
  const int cut = s_cut;
  __syncthreads();

  // Phase C: compact candidates (keys overwrites hist; only `cut` survives)
  for (int i = tid; i < CAP; i += 1024) keys[i] = 0ULL;
  __syncthreads();
  for (int t0 = 0; t0 < HWN; t0 += TILE_F) {
    stage16(sb + t0 + tid * 4, &tile[tid * 4]);
    stage_wait();
    __syncthreads();
#pragma unroll
    for (int j = 0; j < 4; ++j) {
      float s = tile[tid * 4 + j];
      if (s > SCORE_TH) {
        int bin = (int)((s - SCORE_TH) * 20480.0f);
        bin = bin > (NBIN - 1) ? (NBIN - 1) : bin;
        if (bin >= cut) {
          int pos = atomicAdd(&s_cnt, 1);
          if (pos < CAP) {
            unsigned int idx = (unsigned int)(t0 + tid * 4 + j);
            keys[pos] = ((unsigned long long)__float_as_uint(s) << 32) |
                        (unsigned long long)(~idx);
          }
        }
      }
    }
    __syncthreads();
  }
  __syncthreads();

  // Phase D: bitonic sort, descending
  for (int ksz = 2; ksz <= CAP; ksz <<= 1) {
    for (int j = ksz >> 1; j > 0; j >>= 1) {
      for (int t = tid; t < CAP; t += 1024) {
        int ixj = t ^ j;
        if (ixj > t) {
          unsigned long long a = keys[t];
          unsigned long long c = keys[ixj];
          bool desc = ((t & ksz) == 0);
          if ((a < c) == desc) { keys[t] = c; keys[ixj] = a; }
        }
      }
      __syncthreads();
    }
  }

  // Phase E
  {
    unsigned long long key = keys[tid];
    unsigned int idx = (key == 0ULL) ? 0u : ~(unsigned int)(key & 0xFFFFFFFFu);
    topIdx[b * TOPK + tid] = idx;
    topScore[b * TOPK + tid] = __uint_as_float((unsigned int)(key >> 32));
  }
}

// =====================================================================
// Kernel 2: per-candidate rotated-box restoration + AABB + quad output
// =====================================================================
__global__ __launch_bounds__(256) void quads_kernel(
    const float* __restrict__ geo, const unsigned int* __restrict__ topIdx,
    float* __restrict__ out, float* __restrict__ bx0, float* __restrict__ bx1,
    float* __restrict__ by0, float* __restrict__ by1, float* __restrict__ bar) {
  int g = blockIdx.x * blockDim.x + threadIdx.x;
  if (g >= BATCH * TOPK) return;
  int b = g >> 10;
  unsigned int idx = topIdx[g];
  const float* gp = geo + ((size_t)b * HWN + idx) * 5;
  float d0 = gp[0], d1 = gp[1], d2 = gp[2], d3 = gp[3], ang = gp[4];
  float h = d0 + d2, w = d1 + d3;
  float ca = cosf(ang), sa = sinf(ang);

  float px[5], py[5];
  if (ang >= 0.0f) {
    px[0] = 0.f;  py[0] = -h;  px[1] = w;    py[1] = -h;
    px[2] = w;    py[2] = 0.f; px[3] = 0.f;  py[3] = 0.f;
    px[4] = d3;   py[4] = -d0;
  } else {
    px[0] = -w;   py[0] = -h;  px[1] = 0.f;  py[1] = -h;
    px[2] = 0.f;  py[2] = 0.f; px[3] = -w;   py[3] = 0.f;
    px[4] = -d1;  py[4] = -d0;
  }
  float ox = (float)(idx % WW) * 4.0f;
  float oy = (float)(idx / WW) * 4.0f;

  float rxv[5], ryv[5];
#pragma unroll
  for (int j = 0; j < 5; ++j) {
    rxv[j] = px[j] * ca + py[j] * sa;
    ryv[j] = -px[j] * sa + py[j] * ca;
  }
  float tx = ox - rxv[4], ty = oy - ryv[4];

  float X0 = 1e30f, X1 = -1e30f, Y0 = 1e30f, Y1 = -1e30f;
  float* o = out + (size_t)g * 9;
#pragma unroll
  for (int j = 0; j < 4; ++j) {
    float qx = rxv[j] + tx, qy = ryv[j] + ty;
    o[2 * j] = qx;
    o[2 * j + 1] = qy;
    X0 = fminf(X0, qx); X1 = fmaxf(X1, qx);
    Y0 = fminf(Y0, qy); Y1 = fmaxf(Y1, qy);
  }
  bx0[g] = X0; bx1[g] = X1; by0[g] = Y0; by1[g] = Y1;
  bar[g] = fmaxf(X1 - X0, 0.f) * fmaxf(Y1 - Y0, 0.f);
}

// =====================================================================
// Kernel 3: one wave per candidate — direct rectangular box-sum (== SAT diff)
// Row-outer / lane-strided-column-inner: no integer div/mod in the loop.
// =====================================================================
__global__ __launch_bounds__(256) void boxsum_kernel(
    const float* __restrict__ score, const float* __restrict__ bx0,
    const float* __restrict__ bx1, const float* __restrict__ by0,
    const float* __restrict__ by1, float* __restrict__ mean) {
  int wid = (blockIdx.x * blockDim.x + threadIdx.x) >> 5;
  int lane = threadIdx.x & 31;
  if (wid >= BATCH * TOPK) return;
  int b = wid >> 10;
  float X0 = bx0[wid], X1 = bx1[wid], Y0 = by0[wid], Y1 = by1[wid];
  int cx0 = min(max((int)(X0 * 0.25f), 0), WW);
  int cx1 = min(max((int)(X1 * 0.25f) + 1, 0), WW);
  int cy0 = min(max((int)(Y0 * 0.25f), 0), HH);
  int cy1 = min(max((int)(Y1 * 0.25f) + 1, 0), HH);
  const float* sb = score + (size_t)b * HWN;
  float s = 0.f;
  for (int yy = cy0; yy < cy1; ++yy) {
    const float* row = sb + yy * WW;
    for (int xx = cx0 + lane; xx < cx1; xx += 32) s += row[xx];
  }
#pragma unroll
  for (int off = 16; off > 0; off >>= 1) s += __shfl_down(s, off);
  if (lane == 0) {
    int n = (cx1 - cx0) * (cy1 - cy0);
    mean[wid] = s / (float)max(n, 1);
  }
}

// =====================================================================
// Kernel 4: per-batch NMS.
//  - 16x16 IoU tiles: union term area_i+area_j via one f32 WMMA (rank-2
//    outer-sum, co-executes with VALU intersection math); the iou>thresh
//    test is algebraically rewritten as inter > t*(union+eps) (union+eps > 0
//    always), eliminating the IEEE division chain from the hot loop.
//  - predicate ballots packed into a 1024x1024 bitmask in workspace
//  - chunked serial bit-sweep for greedy NMS, then mean-score gate
// =====================================================================
__global__ __launch_bounds__(1024) void nms_kernel(
    const float* __restrict__ topScore, const float* __restrict__ bx0,
    const float* __restrict__ bx1, const float* __restrict__ by0,
    const float* __restrict__ by1, const float* __restrict__ bar,
    const float* __restrict__ mean, unsigned int* __restrict__ mask,
    float* __restrict__ out) {
  const int b = blockIdx.x;
  const int tid = threadIdx.x;
  const int lane = tid & 31;
  const int wv = tid >> 5;
  __shared__ float X0[TOPK], X1[TOPK], Y0[TOPK], Y1[TOPK], AR[TOPK];
  __shared__ unsigned int keepW[32];
  __shared__ unsigned int rowbuf[32][32];
  unsigned int* mb = mask + (size_t)b * TOPK * 32;
  const int base = b * TOPK;

  X0[tid] = bx0[base + tid];
  X1[tid] = bx1[base + tid];
  Y0[tid] = by0[base + tid];
  Y1[tid] = by1[base + tid];
  AR[tid] = bar[base + tid];
  for (int i = tid; i < TOPK * 32; i += 1024) mb[i] = 0u;
  __syncthreads();

  // ---- suppression-mask tiles: 64x64 tiles of 16x16, 32 waves round-robin
  for (int t = wv; t < 64 * 64; t += 32) {
    const int ti = t >> 6, tj = t & 63;
    const int hi = lane >> 4, ln = lane & 15;

    // WMMA operands: A(16x4)=[area_row, 1, 0, 0], B(4x16)=[1; area_colT; 0; 0]
    v2f a, bb;
    a.x = hi ? 0.f : AR[ti * 16 + ln];   // K=0 (lo lanes) / K=2 (hi lanes)
    a.y = hi ? 0.f : 1.0f;               // K=1 / K=3
    bb.x = hi ? 0.f : 1.0f;              // row K=0 / K=2
    bb.y = hi ? 0.f : AR[tj * 16 + ln];  // row K=1 / K=3
    float sumv[8];
#if defined(HAVE_WMMA_F32)
    v8f c = {};
    v8f d = __builtin_amdgcn_wmma_f32_16x16x4_f32(false, a, false, bb,
                                                  (short)0, c, false, false);
#pragma unroll
    for (int v = 0; v < 8; ++v) sumv[v] = d[v];
#else
#pragma unroll
    for (int v = 0; v < 8; ++v)
      sumv[v] = AR[ti * 16 + hi * 8 + v] + AR[tj * 16 + ln];
#endif

    const int j = tj * 16 + ln;   // lane's column (C/D layout: N = lane&15)
    const float jx0 = X0[j], jx1 = X1[j], jy0 = Y0[j], jy1 = Y1[j];
#pragma unroll
    for (int v = 0; v < 8; ++v) {
      const int i = ti * 16 + hi * 8 + v;   // lane's row for VGPR v
      float iw = fminf(X1[i], jx1) - fmaxf(X0[i], jx0);
      float ih = fminf(Y1[i], jy1) - fmaxf(Y0[i], jy0);
      float inter = fmaxf(iw, 0.f) * fmaxf(ih, 0.f);
      // iou > t  <=>  inter > t * (union + eps), union + eps strictly > 0
      bool pred = (inter > NMS_TH * (sumv[v] - inter + 1e-8f)) && (j > i);
      unsigned long long bal = __ballot(pred);
      if (lane == 0) {
        unsigned int lo16 = (unsigned int)bal & 0xFFFFu;
        unsigned int hi16 = ((unsigned int)bal >> 16) & 0xFFFFu;
        int r0 = ti * 16 + v;
        int wi = tj >> 1, sh = (tj & 1) * 16;
        if (lo16) atomicOr(&mb[r0 * 32 + wi], lo16 << sh);
        if (hi16) atomicOr(&mb[(r0 + 8) * 32 + wi], hi16 << sh);
      }
    }
  }
  __syncthreads();

  // ---- keep init from validity
  {
    bool valid = topScore[base + tid] > SCORE_TH;
    unsigned long long bal = __ballot(valid);
    if (lane == 0) keepW[wv] = (unsigned int)bal;
  }
  __syncthreads();

  // ---- greedy sequential NMS, 32-row chunks staged through LDS
  for (int ch = 0; ch < 32; ++ch) {
    rowbuf[wv][lane] = mb[(ch * 32 + wv) * 32 + lane];
    __syncthreads();
    if (tid == 0) {
      for (int ii = 0; ii < 32; ++ii) {
        int i = ch * 32 + ii;
        if ((keepW[i >> 5] >> (i & 31)) & 1u) {
          for (int w2 = 0; w2 < 32; ++w2) keepW[w2] &= ~rowbuf[ii][w2];
        }
      }
    }
    __syncthreads();
  }

  // ---- finalize
  float m = mean[base + tid];
  bool kept = (keepW[tid >> 5] >> (tid & 31)) & 1u;
  out[(size_t)(base + tid) * 9 + 8] = (kept && m > BOX_TH) ? m : 0.0f;
}

// =====================================================================
extern "C" void kernel_launch(void* const* d_in, const int* in_sizes, int n_in,
                              void* d_out, int out_size, void* d_ws,
                              size_t ws_size, hipStream_t stream) {
  const float* score = (const float*)d_in[0];   // (8,640,640,1) f32
  const float* geo = (const float*)d_in[1];     // (8,640,640,5) f32
  float* out = (float*)d_out;                   // (8,1024,9) f32
  (void)in_sizes; (void)n_in; (void)out_size; (void)ws_size;

  uint8_t* w = (uint8_t*)d_ws;
  unsigned int* topIdx = (unsigned int*)w; w += (size_t)BATCH * TOPK * 4;
  float* topScore = (float*)w;             w += (size_t)BATCH * TOPK * 4;
  float* bx0 = (float*)w;                  w += (size_t)BATCH * TOPK * 4;
  float* bx1 = (float*)w;                  w += (size_t)BATCH * TOPK * 4;
  float* by0 = (float*)w;                  w += (size_t)BATCH * TOPK * 4;
  float* by1 = (float*)w;                  w += (size_t)BATCH * TOPK * 4;
  float* bar = (float*)w;                  w += (size_t)BATCH * TOPK * 4;
  float* mean = (float*)w;                 w += (size_t)BATCH * TOPK * 4;
  unsigned int* mask = (unsigned int*)w;   // BATCH*TOPK*32 u32 = 1 MB

  select_topk_kernel<<<BATCH, 1024, 0, stream>>>(score, topIdx, topScore);
  quads_kernel<<<(BATCH * TOPK) / 256, 256, 0, stream>>>(
      geo, topIdx, out, bx0, bx1, by0, by1, bar);
  boxsum_kernel<<<(BATCH * TOPK) / 8, 256, 0, stream>>>(
      score, bx0, bx1, by0, by1, mean);
  nms_kernel<<<BATCH, 1024, 0, stream>>>(topScore, bx0, bx1, by0, by1, bar,
                                         mean, mask, out);
}